// FaceKernelCorrelation_34325378630094
// MI455X (gfx1250) — compile-verified
//
#include <hip/hip_runtime.h>

// FaceKernelCorrelation on gfx1250 (MI455X).
// Phase 1: gather fea points; WMMA f32 16x16x4 computes the ENTIRE exp2
//          argument (C2*dot + C1*(|fea|^2 + |w|^2) - 4) by folding the
//          row-dependent additive term into the spare K=3 contraction slot.
//          Per element: one v_exp_f32. DPP8 quad reductions, kc -> d_out,
//          per-block BN partials -> d_ws.
// Phase 2: deterministic reduction of partials -> mean/var per channel.
// Phase 3: in-place BatchNorm + ReLU on d_out.

#define B_ 8
#define N_ 16384
#define K_ 64
#define BN_EPS 1e-5f

// exp(-12.5*x) = exp2(C1*x);  exp(+25*d) = exp2(C2*d);  "-4" folds the /16.
#define C1_ (-18.0336800f)   // -12.5 * log2(e)
#define C2_ ( 36.0673601f)   //  25.0 * log2(e)
#define CROW_ (C1_ - 4.0f)   // C1*|w|^2 (=1) - 4, folded into the A pad slot

#define NBLK 256
#define NTHR 256
#define WAVES_PER_BLK 8
#define GROUPS (B_ * N_ / 4)                       // 32768 groups of 4 faces
#define TOTAL_WAVES (NBLK * WAVES_PER_BLK)         // 2048
#define GROUPS_PER_WAVE (GROUPS / TOTAL_WAVES)     // 16

typedef __attribute__((ext_vector_type(2))) float v2f;
typedef __attribute__((ext_vector_type(8))) float v8f;

// add value from lane^1 within each quad (DPP8, sel = [1,0,3,2,5,4,7,6])
__device__ __forceinline__ float dpp_xor1_add(float v) {
  const int sel = 1 | (0 << 3) | (3 << 6) | (2 << 9) | (5 << 12) | (4 << 15) | (7 << 18) | (6 << 21);
  return v + __int_as_float(__builtin_amdgcn_mov_dpp8(__float_as_int(v), sel));
}

// add value from lane^2 within each quad (DPP8, sel = [2,3,0,1,6,7,4,5])
__device__ __forceinline__ float dpp_xor2_add(float v) {
  const int sel = 2 | (3 << 3) | (0 << 6) | (1 << 9) | (6 << 12) | (7 << 15) | (4 << 18) | (5 << 21);
  return v + __int_as_float(__builtin_amdgcn_mov_dpp8(__float_as_int(v), sel));
}

__global__ __launch_bounds__(NTHR) void fkc_phase1(
    const float* __restrict__ normals,   // [B,3,N]
    const int*   __restrict__ nbr,       // [B,N,3]
    const float* __restrict__ walpha,    // [1,K,4]
    const float* __restrict__ wbeta,     // [1,K,4]
    float* __restrict__ out,             // [B,K,N] (kc before BN)
    float* __restrict__ part)            // [NBLK][K][2] partial sums
{
  const int tid  = threadIdx.x;
  const int lane = tid & 31;
  const int wave = tid >> 5;
  const int h    = lane >> 4;    // half-wave: 0 -> K=(x,y); 1 -> K=(z,pad)
  const int cl   = lane & 15;

  // ---- Precompute B tiles: 256 kernel-point columns = 16 tiles of 16 cols ----
  // B rows: K0..2 = (wx,wy,wz), K3 = 1.0 (carries the A-side additive term).
  v2f Bt[16];
#pragma unroll
  for (int g = 0; g < 16; ++g) {
    int col = (g << 4) | cl;          // column = k*4 + j
    int kk = col >> 2, jj = col & 3;
    float al = walpha[kk * 4 + jj];
    float be = wbeta [kk * 4 + jj];
    float sa = __sinf(al), ca = __cosf(al);
    float sb = __sinf(be), cb = __cosf(be);
    float wx = sa * cb, wy = sa * sb, wz = ca;
    v2f bb;
    bb[0] = h ? wz  : wx;
    bb[1] = h ? 1.f : wy;
    Bt[g] = bb;
  }

  float accs[16], accq[16];
#pragma unroll
  for (int g = 0; g < 16; ++g) { accs[g] = 0.f; accq[g] = 0.f; }

  const int wid = blockIdx.x * WAVES_PER_BLK + wave;
  const int m  = cl;        // point index within group: face*4 + i
  const int fc = m >> 2;    // face 0..3
  const int ii = m & 3;     // 0 = center, 1..3 = neighbors

  for (int it = 0; it < GROUPS_PER_WAVE; ++it) {
    int grp = wid * GROUPS_PER_WAVE + it;
    int b   = grp >> 12;              // 4096 groups per batch
    int nb  = (grp & 4095) << 2;      // base face index
    int n   = nb + fc;
    int idx = (ii == 0) ? n : nbr[(b * N_ + n) * 3 + (ii - 1)];
    const float* np = normals + (size_t)b * 3 * N_;
    float x = np[idx];
    float y = np[N_ + idx];
    float z = np[2 * N_ + idx];
    float fn2 = x * x + y * y + z * z;

    // A rows: (C2*x, C2*y | C2*z, C1*fn2 + C1 - 4)
    v2f A;
    A[0] = h ? C2_ * z : C2_ * x;
    A[1] = h ? __builtin_fmaf(C1_, fn2, CROW_) : C2_ * y;

#pragma unroll
    for (int g = 0; g < 16; ++g) {
      v8f c = {0.f, 0.f, 0.f, 0.f, 0.f, 0.f, 0.f, 0.f};
      // D[r][c] = C2*dot(fea_r, w_c) + C1*(|fea_r|^2 + 1) - 4  (exp2 argument)
      v8f d = __builtin_amdgcn_wmma_f32_16x16x4_f32(
          false, A, false, Bt[g], (short)0, c, false, false);
      // rows v0..3 -> face 2h ; v4..7 -> face 2h+1
      float p0 = (__builtin_amdgcn_exp2f(d[0]) + __builtin_amdgcn_exp2f(d[1])) +
                 (__builtin_amdgcn_exp2f(d[2]) + __builtin_amdgcn_exp2f(d[3]));
      float p1 = (__builtin_amdgcn_exp2f(d[4]) + __builtin_amdgcn_exp2f(d[5])) +
                 (__builtin_amdgcn_exp2f(d[6]) + __builtin_amdgcn_exp2f(d[7]));
      // reduce over the 4 j-columns of each kernel k (adjacent lanes, DPP8)
      p0 = dpp_xor1_add(p0);  p0 = dpp_xor2_add(p0);
      p1 = dpp_xor1_add(p1);  p1 = dpp_xor2_add(p1);

      if ((lane & 3) == 0) {
        int t = cl >> 2;
        int k = (g << 2) | t;
        size_t o = (((size_t)b * K_ + k) << 14) + (size_t)nb + (h << 1);
        out[o]     = p0;   // face 2h   (1/16 already folded via exp2 "-4")
        out[o + 1] = p1;   // face 2h+1
      }

      // BN partials: each half accumulates its own two faces; halves are
      // combined later in the fixed-order LDS reduction (deterministic).
      accs[g] += p0 + p1;
      accq[g] += __builtin_fmaf(p0, p0, p1 * p1);
    }
  }

  __shared__ float lds[WAVES_PER_BLK][2][K_][2];
  if ((lane & 3) == 0) {              // lanes 0,4,...,28 (j==0, both halves)
    int t = cl >> 2;                  // 0..3
#pragma unroll
    for (int g = 0; g < 16; ++g) {
      int k = (g << 2) | t;
      lds[wave][h][k][0] = accs[g];
      lds[wave][h][k][1] = accq[g];
    }
  }
  __syncthreads();
  if (tid < 2 * K_) {
    int k = tid >> 1, comp = tid & 1;
    float s = 0.f;
#pragma unroll
    for (int w = 0; w < WAVES_PER_BLK; ++w) {
      s += lds[w][0][k][comp];
      s += lds[w][1][k][comp];
    }
    part[blockIdx.x * (2 * K_) + tid] = s;
  }
}

__global__ void fkc_phase2(const float* __restrict__ part, float* __restrict__ stats)
{
  int k = threadIdx.x;
  if (k < K_) {
    float s = 0.f, q = 0.f;
    for (int wg = 0; wg < NBLK; ++wg) {        // fixed order -> deterministic
      s += part[wg * (2 * K_) + 2 * k];
      q += part[wg * (2 * K_) + 2 * k + 1];
    }
    const float invn = 1.f / (float)(B_ * N_);
    float mean = s * invn;
    float var  = q * invn - mean * mean;
    stats[2 * k]     = mean;
    stats[2 * k + 1] = var;
  }
}

__global__ __launch_bounds__(256) void fkc_phase3(
    float* __restrict__ out, const float* __restrict__ stats,
    const float* __restrict__ gamma, const float* __restrict__ beta)
{
  const int total = B_ * K_ * N_;
  for (int idx = blockIdx.x * blockDim.x + threadIdx.x; idx < total;
       idx += gridDim.x * blockDim.x) {
    int k = (idx >> 14) & (K_ - 1);
    float x    = out[idx];
    float mean = stats[2 * k];
    float var  = stats[2 * k + 1];
    float y = gamma[k] * (x - mean) * rsqrtf(var + BN_EPS) + beta[k];
    out[idx] = fmaxf(y, 0.f);
  }
}

extern "C" void kernel_launch(void* const* d_in, const int* in_sizes, int n_in,
                              void* d_out, int out_size, void* d_ws, size_t ws_size,
                              hipStream_t stream) {
  const float* normals = (const float*)d_in[0];
  const int*   nbr     = (const int*)  d_in[1];
  const float* walpha  = (const float*)d_in[2];
  const float* wbeta   = (const float*)d_in[3];
  const float* gamma   = (const float*)d_in[4];
  const float* beta    = (const float*)d_in[5];
  float* out   = (float*)d_out;
  float* part  = (float*)d_ws;            // NBLK * 2K floats = 128 KB
  float* stats = part + NBLK * 2 * K_;    // 2K floats

  hipLaunchKernelGGL(fkc_phase1, dim3(NBLK), dim3(NTHR), 0, stream,
                     normals, nbr, walpha, wbeta, out, part);
  hipLaunchKernelGGL(fkc_phase2, dim3(1), dim3(64), 0, stream, part, stats);
  hipLaunchKernelGGL(fkc_phase3, dim3(2048), dim3(256), 0, stream,
                     out, stats, gamma, beta);
}